// AAGCN_55731495633519
// MI455X (gfx1250) — compile-verified
//
#include <hip/hip_runtime.h>
#include <math.h>

typedef float v2f __attribute__((ext_vector_type(2)));
typedef float v8f __attribute__((ext_vector_type(8)));

constexpr int N = 32, C = 64, T = 512, V = 21, E = 16;
constexpr float EPS = 1e-5f;

// ---------------- K1: BN1 with the raw-reshape permutation ----------------
// xb[n,c',t',v] = BN_{j=v*64+c'}( x[n, t'&63, c'*8 + (t'>>6), v] )
__global__ void k_bn1(const float* __restrict__ x,
                      const float* __restrict__ g, const float* __restrict__ b,
                      const float* __restrict__ mu, const float* __restrict__ var,
                      float* __restrict__ xb) {
  int idx = blockIdx.x * blockDim.x + threadIdx.x;
  const int total = N * C * T * V;
  if (idx >= total) return;
  int v  = idx % V;
  int t1 = (idx / V) % T;
  int c1 = (idx / (V * T)) % C;
  int n  = idx / (V * T * C);
  int cin = t1 & 63;
  int tin = (c1 << 3) + (t1 >> 6);
  int j = v * C + c1;
  size_t in_idx = (((size_t)n * C + cin) * T + tin) * V + v;
  float sc = g[j] * rsqrtf(var[j] + EPS);
  xb[idx] = (x[in_idx] - mu[j]) * sc + b[j];
}

// ---------------- K2: fused attention partial sums ----------------
// Per block: one (n, s, t-chunk of 64). Accumulates sum_{e,t} a1[i,e,t]*a2[e,t,j].
__global__ void __launch_bounds__(256) k_attn_partial(
    const float* __restrict__ xb,
    const float* __restrict__ wa, const float* __restrict__ ba,
    const float* __restrict__ wb, const float* __restrict__ bb,
    float* __restrict__ part) {
  __shared__ float was[E * C], wbs[E * C];
  __shared__ float xs[C * 4 * V];   // [c][tt][v], tt-tile of 4
  __shared__ float a1s[4 * V * E];  // [tt][v][e]
  __shared__ float a2s[4 * E * V];  // [tt][e][v]
  int bid = blockIdx.x;
  int n = bid / 24;
  int r = bid % 24;
  int s = r / 8;
  int chunk = r % 8;
  int tid = threadIdx.x;
  for (int i = tid; i < E * C; i += 256) {
    was[i] = wa[s * E * C + i];
    wbs[i] = wb[s * E * C + i];
  }
  float acc0 = 0.f, acc1 = 0.f;
  int p0 = tid, p1 = tid + 256;
  const float* xbn = xb + (size_t)n * C * T * V;
  for (int tile = 0; tile < 16; ++tile) {
    int t0 = chunk * 64 + tile * 4;
    __syncthreads();  // prior accumulate done; was/wbs ready on iter 0
    for (int i = tid; i < C * 4 * V; i += 256) {
      int c = i / 84;
      int rem = i % 84;
      int tt = rem / 21;
      int w = rem % 21;
      xs[i] = xbn[((size_t)c * T + t0 + tt) * V + w];
    }
    __syncthreads();
    for (int i = tid; i < 2 * 4 * V * E; i += 256) {
      if (i < 4 * V * E) {
        int tt = i / (V * E);
        int rem = i % (V * E);
        int vv = rem / E;
        int e = rem % E;
        float a = ba[s * E + e];
        const float* xr = &xs[tt * 21 + vv];
        const float* wr = &was[e * C];
        for (int c = 0; c < C; ++c) a += xr[c * 84] * wr[c];
        a1s[i] = a;
      } else {
        int ii = i - 4 * V * E;
        int tt = ii / (E * V);
        int rem = ii % (E * V);
        int e = rem / V;
        int vv = rem % V;
        float a = bb[s * E + e];
        const float* xr = &xs[tt * 21 + vv];
        const float* wr = &wbs[e * C];
        for (int c = 0; c < C; ++c) a += xr[c * 84] * wr[c];
        a2s[ii] = a;
      }
    }
    __syncthreads();
    if (p0 < V * V) {
      int i = p0 / V, jj = p0 % V;
      float a = 0.f;
      for (int tt = 0; tt < 4; ++tt)
        for (int e = 0; e < E; ++e)
          a += a1s[tt * 336 + i * E + e] * a2s[tt * 336 + e * V + jj];
      acc0 += a;
    }
    if (p1 < V * V) {
      int i = p1 / V, jj = p1 % V;
      float a = 0.f;
      for (int tt = 0; tt < 4; ++tt)
        for (int e = 0; e < E; ++e)
          a += a1s[tt * 336 + i * E + e] * a2s[tt * 336 + e * V + jj];
      acc1 += a;
    }
  }
  if (p0 < V * V) part[(size_t)bid * 441 + p0] = acc0;
  if (p1 < V * V) part[(size_t)bid * 441 + p1] = acc1;
}

// ---------------- K2b: reduce chunks, tanh, build adjacency ----------------
__global__ void k_attn_final(const float* __restrict__ part,
                             const float* __restrict__ PA,
                             const float* __restrict__ alpha,
                             float* __restrict__ adjb) {
  int idx = blockIdx.x * blockDim.x + threadIdx.x;
  if (idx >= N * 3 * 441) return;
  int p = idx % 441;
  int s = (idx / 441) % 3;
  int n = idx / 1323;
  float sum = 0.f;
  for (int ch = 0; ch < 8; ++ch)
    sum += part[(size_t)(n * 24 + s * 8 + ch) * 441 + p];
  float attn = tanhf(sum * (1.0f / 8192.0f));
  adjb[idx] = PA[s * 441 + p] + attn * alpha[0];
}

// ---------------- K3: WMMA GEMM: y = sum_s wd[s] x (xb . adj[s]) + BN2 + res + relu
// Block = (n, 8-wide t tile). 11 waves, wave nj owns column tile nj (j = tt*21+v).
__global__ void __launch_bounds__(352) k_ygemm(
    const float* __restrict__ xb, const float* __restrict__ adjb,
    const float* __restrict__ wd, const float* __restrict__ bd,
    const float* __restrict__ g2, const float* __restrict__ b2,
    const float* __restrict__ mu2, const float* __restrict__ var2,
    float* __restrict__ y0) {
  __shared__ float xs[C * 8 * V];  // [c][tt][w] : 64*168 floats (43 KB)
  __shared__ float adjs[3 * 441];
  int n = blockIdx.x >> 6;
  int t0 = (blockIdx.x & 63) * 8;
  int tid = threadIdx.x;
  int wave = tid >> 5;
  int lane = tid & 31;
  const float* xbn = xb + (size_t)n * C * T * V;
  for (int i = tid; i < C * 168; i += 352) {
    int c = i / 168;
    int rem = i % 168;
    int tt = rem / 21;
    int w = rem % 21;
    xs[i] = xbn[((size_t)c * T + t0 + tt) * V + w];
  }
  for (int i = tid; i < 3 * 441; i += 352) adjs[i] = adjb[(size_t)n * 3 * 441 + i];
  __syncthreads();

  int nj = wave;              // 0..10 column tiles (168 cols padded to 176)
  int jl = lane & 15;
  int j = nj * 16 + jl;
  bool jv = (j < 168);
  int tt = jv ? (j / 21) : 0;
  int vv = jv ? (j % 21) : 0;
  int off = (lane < 16) ? 0 : 2;

  const v8f vzero = {0.f, 0.f, 0.f, 0.f, 0.f, 0.f, 0.f, 0.f};
  v8f acc[4];
  acc[0] = vzero; acc[1] = vzero; acc[2] = vzero; acc[3] = vzero;

  for (int s = 0; s < 3; ++s) {
    // Per-lane B fragments: Q[c][j] = sum_w xs[c][tt][w] * adj[s][w][vv]
    float qreg[32];
    if (jv) {
      const float* aj = &adjs[s * 441 + vv];
#pragma unroll
      for (int st = 0; st < 16; ++st) {
        int c0 = 4 * st + off;
        const float* x0 = &xs[c0 * 168 + tt * 21];
        const float* x1 = x0 + 168;
        float s0 = 0.f, s1 = 0.f;
        for (int w = 0; w < 21; ++w) {
          float aw = aj[w * 21];
          s0 += x0[w] * aw;
          s1 += x1[w] * aw;
        }
        qreg[2 * st] = s0;
        qreg[2 * st + 1] = s1;
      }
    } else {
#pragma unroll
      for (int q = 0; q < 32; ++q) qreg[q] = 0.f;
    }
    // A-fragment base: row m = mi*16 + (lane%16), cols kk = 4*st + off.
    // Batch all 16 A-fragments of an M-tile into registers (16 independent
    // global_load_b64, 8B-aligned), then issue 16 back-to-back WMMAs.
    const float* wds = wd + s * 4096 + jl * 64 + off;
#pragma unroll
    for (int mi = 0; mi < 4; ++mi) {
      const float* wrow = wds + mi * 16 * 64;
      v2f af[16];
#pragma unroll
      for (int st = 0; st < 16; ++st) {
        af[st] = *(const v2f*)(wrow + 4 * st);
      }
#pragma unroll
      for (int st = 0; st < 16; ++st) {
        v2f bfr;
        bfr.x = qreg[2 * st];
        bfr.y = qreg[2 * st + 1];
        acc[mi] = __builtin_amdgcn_wmma_f32_16x16x4_f32(
            false, af[st], false, bfr, (short)0, acc[mi], false, false);
      }
    }
  }
  if (!jv) return;
  int t = t0 + tt;
#pragma unroll
  for (int mi = 0; mi < 4; ++mi) {
#pragma unroll
    for (int rr = 0; rr < 8; ++rr) {
      int o = mi * 16 + rr + ((lane < 16) ? 0 : 8);
      float gv = acc[mi][rr] + (bd[o] + bd[64 + o] + bd[128 + o]);
      float sc = g2[o] * rsqrtf(var2[o] + EPS);
      gv = (gv - mu2[o]) * sc + b2[o];
      gv += xs[o * 168 + tt * 21 + vv];  // residual xb
      y0[(((size_t)n * C + o) * T + t) * V + vv] = fmaxf(gv, 0.f);
    }
  }
}

// ---------------- SE-gate chain (collapsed to multiplicative factors) ------
__global__ void k_ses(const float* __restrict__ y0, float* __restrict__ ses) {
  int idx = blockIdx.x * blockDim.x + threadIdx.x;
  if (idx >= N * C * V) return;
  int v = idx % V;
  int c = (idx / V) % C;
  int n = idx / (V * C);
  const float* yp = y0 + ((size_t)(n * C + c)) * T * V + v;
  float s = 0.f;
  for (int t = 0; t < T; ++t) s += yp[(size_t)t * V];
  ses[idx] = s * (1.f / T);
}

__global__ void k_sa(const float* __restrict__ ses, const float* __restrict__ w,
                     const float* __restrict__ bsa, float* __restrict__ sa) {
  int idx = blockIdx.x * blockDim.x + threadIdx.x;
  if (idx >= N * V) return;
  int vo = idx % V;
  int n = idx / V;
  float a = bsa[0];
  for (int c = 0; c < C; ++c) {
    const float* sp = ses + (n * C + c) * V;
    const float* wp = w + c * V;
    for (int k = 0; k < V; ++k) {
      int vi = vo + k - 10;
      if (vi >= 0 && vi < V) a += sp[vi] * wp[k];
    }
  }
  sa[idx] = 1.f / (1.f + expf(-a));
}

__global__ void k_set(const float* __restrict__ y0, const float* __restrict__ sa,
                      float* __restrict__ setb) {
  int idx = blockIdx.x * blockDim.x + threadIdx.x;
  if (idx >= N * C * T) return;
  int t = idx % T;
  int c = (idx / T) % C;
  int n = idx / (T * C);
  const float* yp = y0 + (((size_t)(n * C + c)) * T + t) * V;
  const float* sp = sa + n * V;
  float s = 0.f;
  for (int v = 0; v < V; ++v) s += yp[v] * (1.f + sp[v]);
  setb[idx] = s * (1.f / V);
}

__global__ void k_ta(const float* __restrict__ setb, const float* __restrict__ w,
                     const float* __restrict__ bta, float* __restrict__ ta) {
  int idx = blockIdx.x * blockDim.x + threadIdx.x;
  if (idx >= N * T) return;
  int t = idx % T;
  int n = idx / T;
  float a = bta[0];
  for (int c = 0; c < C; ++c) {
    const float* sp = setb + (size_t)(n * C + c) * T;
    const float* wp = w + c * 9;
    for (int k = 0; k < 9; ++k) {
      int ti = t + k - 4;
      if (ti >= 0 && ti < T) a += sp[ti] * wp[k];
    }
  }
  ta[idx] = 1.f / (1.f + expf(-a));
}

__global__ void __launch_bounds__(256) k_sec(const float* __restrict__ y0,
                                             const float* __restrict__ sa,
                                             const float* __restrict__ ta,
                                             float* __restrict__ sec) {
  __shared__ float red[256];
  int b = blockIdx.x;
  int n = b >> 6;
  int c = b & 63;
  int tid = threadIdx.x;
  const float* yp = y0 + ((size_t)(n * C + c)) * T * V;
  float s = 0.f;
  for (int i = tid; i < T * V; i += 256) {
    int t = i / V, v = i % V;
    s += yp[i] * (1.f + sa[n * V + v]) * (1.f + ta[n * T + t]);
  }
  red[tid] = s;
  __syncthreads();
  for (int st = 128; st > 0; st >>= 1) {
    if (tid < st) red[tid] += red[tid + st];
    __syncthreads();
  }
  if (tid == 0) sec[b] = red[0] * (1.f / (float)(T * V));
}

__global__ void k_ca(const float* __restrict__ sec, const float* __restrict__ w1,
                     const float* __restrict__ bf1, const float* __restrict__ w2,
                     const float* __restrict__ bf2, float* __restrict__ ca) {
  __shared__ float sl[64];
  __shared__ float h[32];
  int n = blockIdx.x;
  int tid = threadIdx.x;
  sl[tid] = sec[n * 64 + tid];
  __syncthreads();
  if (tid < 32) {
    float a = bf1[tid];
    for (int c = 0; c < 64; ++c) a += sl[c] * w1[tid * 64 + c];
    h[tid] = fmaxf(a, 0.f);
  }
  __syncthreads();
  float a = bf2[tid];
  for (int m = 0; m < 32; ++m) a += h[m] * w2[tid * 32 + m];
  ca[n * 64 + tid] = 1.f / (1.f + expf(-a));
}

__global__ void k_final(const float* __restrict__ y0, const float* __restrict__ sa,
                        const float* __restrict__ ta, const float* __restrict__ ca,
                        float* __restrict__ out) {
  int idx = blockIdx.x * blockDim.x + threadIdx.x;
  const int total = N * C * T * V;
  if (idx >= total) return;
  int v = idx % V;
  int t = (idx / V) % T;
  int c = (idx / (V * T)) % C;
  int n = idx / (V * T * C);
  float f = y0[idx] * (1.f + sa[n * V + v]) * (1.f + ta[n * T + t]) *
            (1.f + ca[n * C + c]);
  out[idx] = fmaxf(f, 0.f);
}

extern "C" void kernel_launch(void* const* d_in, const int* in_sizes, int n_in,
                              void* d_out, int out_size, void* d_ws, size_t ws_size,
                              hipStream_t stream) {
  (void)in_sizes; (void)n_in; (void)out_size; (void)ws_size;
  const float* x   = (const float*)d_in[0];
  const float* g1  = (const float*)d_in[2];
  const float* b1  = (const float*)d_in[3];
  const float* m1  = (const float*)d_in[4];
  const float* v1  = (const float*)d_in[5];
  const float* PA  = (const float*)d_in[6];
  const float* al  = (const float*)d_in[7];
  const float* wa  = (const float*)d_in[8];
  const float* ba  = (const float*)d_in[9];
  const float* wb  = (const float*)d_in[10];
  const float* bb  = (const float*)d_in[11];
  const float* wd  = (const float*)d_in[12];
  const float* bd  = (const float*)d_in[13];
  const float* g2  = (const float*)d_in[14];
  const float* b2  = (const float*)d_in[15];
  const float* m2  = (const float*)d_in[16];
  const float* v2  = (const float*)d_in[17];
  const float* wsa = (const float*)d_in[18];
  const float* bsa = (const float*)d_in[19];
  const float* wta = (const float*)d_in[20];
  const float* bta = (const float*)d_in[21];
  const float* w1  = (const float*)d_in[22];
  const float* bf1 = (const float*)d_in[23];
  const float* w2  = (const float*)d_in[24];
  const float* bf2 = (const float*)d_in[25];
  float* out = (float*)d_out;
  float* wsf = (float*)d_ws;

  const size_t SZ = (size_t)N * C * T * V;  // 22,020,096
  float* xb    = wsf;
  float* y0    = xb + SZ;
  float* adjb  = y0 + SZ;                       // N*3*441
  float* part  = adjb + (size_t)N * 3 * 441;    // 768*441
  float* ses   = part + (size_t)768 * 441;      // N*C*V
  float* sabuf = ses + (size_t)N * C * V;       // N*V
  float* setb  = sabuf + (size_t)N * V;         // N*C*T
  float* tabuf = setb + (size_t)N * C * T;      // N*T
  float* secb  = tabuf + (size_t)N * T;         // N*C
  float* cabuf = secb + (size_t)N * C;          // N*C

  k_bn1<<<(int)((SZ + 255) / 256), 256, 0, stream>>>(x, g1, b1, m1, v1, xb);
  k_attn_partial<<<N * 24, 256, 0, stream>>>(xb, wa, ba, wb, bb, part);
  k_attn_final<<<(N * 3 * 441 + 255) / 256, 256, 0, stream>>>(part, PA, al, adjb);
  k_ygemm<<<N * 64, 352, 0, stream>>>(xb, adjb, wd, bd, g2, b2, m2, v2, y0);
  k_ses<<<(N * C * V + 255) / 256, 256, 0, stream>>>(y0, ses);
  k_sa<<<(N * V + 255) / 256, 256, 0, stream>>>(ses, wsa, bsa, sabuf);
  k_set<<<(N * C * T + 255) / 256, 256, 0, stream>>>(y0, sabuf, setb);
  k_ta<<<(N * T + 255) / 256, 256, 0, stream>>>(setb, wta, bta, tabuf);
  k_sec<<<N * C, 256, 0, stream>>>(y0, sabuf, tabuf, secb);
  k_ca<<<N, 64, 0, stream>>>(secb, w1, bf1, w2, bf2, cabuf);
  k_final<<<(int)((SZ + 255) / 256), 256, 0, stream>>>(y0, sabuf, tabuf, cabuf, out);
}